// GraphConv_55336358641765
// MI455X (gfx1250) — compile-verified
//
#include <hip/hip_runtime.h>

typedef __attribute__((ext_vector_type(2))) float v2f;
typedef __attribute__((ext_vector_type(8))) float v8f;

#define NUM_NODES      5000
#define NUM_EDGES      80000
#define ELEMS_PER_NODE 2048            // BATCH*SEQ*IN_FEAT = 32*64*1
#define OUT_FEAT       10
#define TOTAL_ELEMS    (NUM_NODES * ELEMS_PER_NODE)   // 10,240,000
#define TILES_PER_BLK  8               // 8 wave32 per 256-thread block
#define ROWS_PER_BLK   (TILES_PER_BLK * 16)           // 128
#define FLOATS_PER_BLK (ROWS_PER_BLK * 2 * OUT_FEAT)  // 2560

// ---------------------------------------------------------------------------
// Kernel 1: zero counts + cursor (harness poisons d_ws).
// ---------------------------------------------------------------------------
__global__ void zero_int_kernel(int* __restrict__ p, int n) {
  int i = blockIdx.x * blockDim.x + threadIdx.x;
  if (i < n) p[i] = 0;
}

// ---------------------------------------------------------------------------
// Kernel 2: in-degree histogram (cheap: 80K int atomics total).
// ---------------------------------------------------------------------------
__global__ void count_kernel(const int* __restrict__ edst,
                             int* __restrict__ counts) {
  int e = blockIdx.x * blockDim.x + threadIdx.x;
  if (e < NUM_EDGES) atomicAdd(&counts[edst[e]], 1);
}

// ---------------------------------------------------------------------------
// Kernel 3: single-block exclusive prefix scan over 5000 counts -> rowstart.
// 256 threads x 20 elements each; Hillis-Steele over per-thread sums in LDS.
// ---------------------------------------------------------------------------
__global__ void scan_kernel(const int* __restrict__ counts,
                            int* __restrict__ rowstart) {
  __shared__ int sums[256];
  const int t = threadIdx.x;
  const int base = t * 20;
  int vals[20];
  int local = 0;
#pragma unroll
  for (int i = 0; i < 20; ++i) {
    const int idx = base + i;
    const int c = (idx < NUM_NODES) ? counts[idx] : 0;
    vals[i] = c;
    local += c;
  }
  sums[t] = local;
  __syncthreads();
  for (int off = 1; off < 256; off <<= 1) {
    const int v = sums[t];
    const int add = (t >= off) ? sums[t - off] : 0;
    __syncthreads();
    sums[t] = v + add;
    __syncthreads();
  }
  int run = (t > 0) ? sums[t - 1] : 0;     // exclusive prefix of this chunk
#pragma unroll
  for (int i = 0; i < 20; ++i) {
    const int idx = base + i;
    if (idx < NUM_NODES) rowstart[idx] = run;
    run += vals[i];
  }
  if (t == 255) rowstart[NUM_NODES] = sums[255];   // == NUM_EDGES
}

// ---------------------------------------------------------------------------
// Kernel 4: bucket edges into CSR (dst -> list of src). 80K int atomics.
// ---------------------------------------------------------------------------
__global__ void build_csr_kernel(const int* __restrict__ esrc,
                                 const int* __restrict__ edst,
                                 const int* __restrict__ rowstart,
                                 int* __restrict__ cursor,
                                 int* __restrict__ esorted) {
  int e = blockIdx.x * blockDim.x + threadIdx.x;
  if (e < NUM_EDGES) {
    const int d = edst[e];
    const int pos = rowstart[d] + atomicAdd(&cursor[d], 1);
    esorted[pos] = esrc[e];
  }
}

// ---------------------------------------------------------------------------
// Kernel 5: fused gather-mean + V_WMMA_F32_16X16X4_F32 + coalesced store.
// Each wave owns a 16-row tile of flattened (node,b,s); 2048 % 16 == 0 so a
// tile never crosses a node boundary. The per-tile neighbor sum is a gather
// over the node's CSR list (features stay resident in the 192MB L2).
//   A (16x4 f32): lanes 0-15 hold row m in VGPR0 (K=0); everything else zero.
//   B (4x16 f32): VGPR0 lanes 0-9 = weight row at K=0.
//   D (16x16 f32): VGPR r -> M = r (lanes 0-15) / r+8 (lanes 16-31), N = lane%16.
// Results staged in LDS (conflict-free writes), then block-wide b128 stores
// of 10,240 perfectly contiguous bytes -> clean path for the dominant 819MB.
// ---------------------------------------------------------------------------
__global__ void wmma_out_kernel(const float* __restrict__ feat,
                                const float* __restrict__ W,
                                const int* __restrict__ rowstart,
                                const int* __restrict__ esorted,
                                float* __restrict__ out) {
  __shared__ __align__(16) float stage[FLOATS_PER_BLK];

  const int  lane = threadIdx.x & 31;
  const int  wave = threadIdx.x >> 5;
  const long tile = (long)blockIdx.x * TILES_PER_BLK + wave;
  const long m0   = tile * 16;                 // first flattened row of tile
  const int  node = (int)(m0 >> 11);           // 2048 elements per node
  const int  off  = (int)(m0 & 2047);          // slice offset within node

  const int r0 = rowstart[node];
  const int r1 = rowstart[node + 1];
  const float cnt = (float)(r1 - r0);
  const float inv = (cnt > 0.0f) ? (1.0f / cnt) : 1.0f;   // TF-style mean denom

  // Gather-sum incoming neighbor slices. Lanes 16-31 mirror lanes 0-15
  // (branch-free 64B coalesced loads); their results are discarded below.
  const int li = lane & 15;
  float acc = 0.0f;
  for (int e = r0; e < r1; ++e) {
    const int s = esorted[e];
    acc += feat[(long)s * ELEMS_PER_NODE + off + li];
  }

  v2f a_rep = {0.0f, 0.0f};
  v2f a_agg = {0.0f, 0.0f};
  if (lane < 16) {
    a_rep.x = feat[m0 + lane];
    a_agg.x = acc * inv;
  }
  v2f b = {0.0f, 0.0f};
  if (lane < OUT_FEAT) b.x = W[lane];

  v8f c = {};
  // 8 args: (neg_a, A, neg_b, B, c_mod, C, reuse_a, reuse_b)
  v8f d_rep = __builtin_amdgcn_wmma_f32_16x16x4_f32(
      false, a_rep, false, b, (short)0, c, false, false);
  v8f d_agg = __builtin_amdgcn_wmma_f32_16x16x4_f32(
      false, a_agg, false, b, (short)0, c, false, false);

  // Stage D tiles in LDS in the final (row-major, 20-wide) output layout.
  const int col   = lane & 15;
  const int lrow0 = wave * 16 + ((lane >> 4) << 3);   // +8 rows for lanes 16-31
  if (col < OUT_FEAT) {
#pragma unroll
    for (int r = 0; r < 8; ++r) {
      const float vr = d_rep[r];
      const float va = d_agg[r];
      stage[(lrow0 + r) * 20 + col]            = vr > 0.0f ? vr : 0.0f;
      stage[(lrow0 + r) * 20 + OUT_FEAT + col] = va > 0.0f ? va : 0.0f;
    }
  }
  __syncthreads();

  // Block-coalesced b128 stores: 2560 contiguous floats per block.
  const float4* __restrict__ src4 = (const float4*)stage;
  float4* __restrict__ dst4 = (float4*)(out + (long)blockIdx.x * FLOATS_PER_BLK);
  for (int i = threadIdx.x; i < FLOATS_PER_BLK / 4; i += blockDim.x)
    dst4[i] = src4[i];
}

// ---------------------------------------------------------------------------
extern "C" void kernel_launch(void* const* d_in, const int* in_sizes, int n_in,
                              void* d_out, int out_size, void* d_ws, size_t ws_size,
                              hipStream_t stream) {
  const float* feat = (const float*)d_in[0];   // (5000,32,64,1) f32
  const float* W    = (const float*)d_in[1];   // (1,10) f32
  const int*   esrc = (const int*)d_in[2];     // (80000,) i32
  const int*   edst = (const int*)d_in[3];     // (80000,) i32
  float*       out  = (float*)d_out;           // (5000,32,64,20) f32

  // Workspace layout (ints): counts | cursor | rowstart | esorted  (~372 KB)
  int* counts   = (int*)d_ws;
  int* cursor   = counts + NUM_NODES;
  int* rowstart = cursor + NUM_NODES;            // NUM_NODES + 1 entries
  int* esorted  = rowstart + (NUM_NODES + 1);

  zero_int_kernel<<<(2 * NUM_NODES + 255) / 256, 256, 0, stream>>>(counts,
                                                                   2 * NUM_NODES);
  count_kernel<<<(NUM_EDGES + 255) / 256, 256, 0, stream>>>(edst, counts);
  scan_kernel<<<1, 256, 0, stream>>>(counts, rowstart);
  build_csr_kernel<<<(NUM_EDGES + 255) / 256, 256, 0, stream>>>(esrc, edst,
                                                                rowstart, cursor,
                                                                esorted);

  const long tiles  = (long)TOTAL_ELEMS / 16;       // 640,000 wave-tiles
  const long blocks = tiles / TILES_PER_BLK;        // 80,000 blocks
  wmma_out_kernel<<<(int)blocks, 256, 0, stream>>>(feat, W, rowstart, esorted, out);
}